// KWinnersCompetition_32710470926554
// MI455X (gfx1250) — compile-verified
//
#include <hip/hip_runtime.h>

typedef __attribute__((ext_vector_type(2))) float v2f;
typedef __attribute__((ext_vector_type(8))) float v8f;

#define B_N   64
#define C_N   512
#define HW_N  784            // 28*28
#define CHW_N (C_N * HW_N)   // 401408

// ---------------------------------------------------------------------------
// Pass 1: per-(b,p) channel sums via V_WMMA_F32_16X16X4_F32 with A = ones.
// One wave32 handles 16 consecutive spatial positions of one batch image and
// accumulates 128 f32 WMMAs over the 512 channels (full fp32 precision).
// D[m][n] = sum_k B[k][n]; every row of D carries the per-position sum, so we
// read row M=0 (acc[0], lanes 0..15) and store mean = sum / 512.
// ---------------------------------------------------------------------------
__global__ void kwc_mean_wmma(const float* __restrict__ x,
                              float* __restrict__ means) {
  const unsigned gt    = blockIdx.x * blockDim.x + threadIdx.x;
  const unsigned wave  = gt >> 5;        // 3136 waves total = 64 b * 49 tiles
  const unsigned lane  = gt & 31;
  const unsigned b     = wave / 49u;     // 49 tiles of 16 positions per image
  const unsigned tile  = wave - b * 49u;
  const unsigned pbase = tile * 16u;
  const unsigned col   = lane & 15u;     // position within the 16-wide tile
  const unsigned khalf = lane >> 4;      // which channel-pair this half loads

  const float* xp = x + (size_t)b * CHW_N + pbase + col;

  v2f a; a.x = 1.0f; a.y = 1.0f;        // A-matrix: 16x4 of ones
  v8f acc = {};

#pragma unroll 4
  for (int it = 0; it < C_N / 4; ++it) {
    const unsigned c0 = 4u * (unsigned)it + 2u * khalf;
    v2f bv;
    bv.x = xp[(size_t)c0 * HW_N];        // channel c0,   position pbase+col
    bv.y = xp[(size_t)(c0 + 1) * HW_N];  // channel c0+1, position pbase+col
    // D = A(ones) x B + C  -> accumulates per-position channel sums
    acc = __builtin_amdgcn_wmma_f32_16x16x4_f32(
        /*neg_a=*/false, a, /*neg_b=*/false, bv,
        /*c_mod=*/(short)0, acc, /*reuse_a=*/false, /*reuse_b=*/false);
  }

  if (lane < 16u) {
    // acc[0]: M=0 row, N = lane  ->  sum over all 512 channels at position
    means[(size_t)b * HW_N + pbase + lane] = acc[0] * (1.0f / (float)C_N);
  }
}

// ---------------------------------------------------------------------------
// Pass 2: out = relu(x - mean[b,p]).  float4-vectorized stream; x re-read is
// L2-resident (103 MB < 192 MB L2), so HBM sees ~1x read + 1x write total.
// ---------------------------------------------------------------------------
__global__ void kwc_sub_relu(const float* __restrict__ x,
                             const float* __restrict__ means,
                             float* __restrict__ out) {
  const unsigned i   = (blockIdx.x * blockDim.x + threadIdx.x) * 4u;
  const unsigned b   = i / CHW_N;
  const unsigned rem = i - b * CHW_N;
  const unsigned p   = rem % HW_N;       // multiple of 4 (HW_N % 4 == 0)

  const float4 xv = *reinterpret_cast<const float4*>(x + i);
  const float4 mv = *reinterpret_cast<const float4*>(means + (size_t)b * HW_N + p);

  float4 o;
  o.x = fmaxf(xv.x - mv.x, 0.0f);
  o.y = fmaxf(xv.y - mv.y, 0.0f);
  o.z = fmaxf(xv.z - mv.z, 0.0f);
  o.w = fmaxf(xv.w - mv.w, 0.0f);
  *reinterpret_cast<float4*>(out + i) = o;
}

extern "C" void kernel_launch(void* const* d_in, const int* in_sizes, int n_in,
                              void* d_out, int out_size, void* d_ws, size_t ws_size,
                              hipStream_t stream) {
  const float* x     = (const float*)d_in[0];   // [64,512,28,28] fp32
  float*       out   = (float*)d_out;           // same shape, fp32
  float*       means = (float*)d_ws;            // 64*784 floats = ~200 KB

  // Pass 1: 3136 full waves (64 images * 49 tiles), 4 waves per block.
  kwc_mean_wmma<<<dim3(784), dim3(128), 0, stream>>>(x, means);

  // Pass 2: 25,690,112 elements / 4 per thread / 256 per block = 25088 blocks.
  kwc_sub_relu<<<dim3((B_N * CHW_N / 4) / 256), dim3(256), 0, stream>>>(x, means, out);
}